// GmiLoss_soft_38972533244659
// MI455X (gfx1250) — compile-verified
//
#include <hip/hip_runtime.h>

typedef __attribute__((ext_vector_type(2))) float v2f;
typedef __attribute__((ext_vector_type(8))) float v8f;

#define NCOL   256
#define D_ROWS 4096
#define EPSF   1e-7f
#define LOG2EF 1.4426950408889634f

__device__ __forceinline__ float waveReduceSum(float v) {
#pragma unroll
    for (int o = 16; o > 0; o >>= 1) v += __shfl_xor(v, o, 32);
    return v;
}

__device__ __forceinline__ float waveReduceMax(float v) {
#pragma unroll
    for (int o = 16; o > 0; o >>= 1) v = fmaxf(v, __shfl_xor(v, o, 32));
    return v;
}

// One block (256 threads = 8 wave32) per row.
// Computes e_x + e_y - e_xy for this row and writes it to partial[row].
__global__ __launch_bounds__(256) void gmi_row_kernel(const float* __restrict__ x,
                                                      const float* __restrict__ y,
                                                      float* __restrict__ partial) {
    __shared__ float sP[NCOL];
    __shared__ float sQ[NCOL];
    __shared__ float sRedA[8];
    __shared__ float sRedB[8];

    const int tid  = threadIdx.x;
    const int lane = tid & 31;
    const int wave = tid >> 5;
    const int row  = blockIdx.x;

    const float xv = x[row * NCOL + tid];
    const float yv = y[row * NCOL + tid];

    // ---- row max (softmax stabilization) ----
    float mx = waveReduceMax(xv);
    float my = waveReduceMax(yv);
    if (lane == 0) { sRedA[wave] = mx; sRedB[wave] = my; }
    __syncthreads();
    mx = sRedA[0]; my = sRedB[0];
#pragma unroll
    for (int w = 1; w < 8; ++w) { mx = fmaxf(mx, sRedA[w]); my = fmaxf(my, sRedB[w]); }
    __syncthreads();

    // ---- exp and row sum ----
    const float ex = __builtin_amdgcn_exp2f((xv - mx) * LOG2EF);  // exp(xv - mx)
    const float ey = __builtin_amdgcn_exp2f((yv - my) * LOG2EF);
    float sx = waveReduceSum(ex);
    float sy = waveReduceSum(ey);
    if (lane == 0) { sRedA[wave] = sx; sRedB[wave] = sy; }
    __syncthreads();
    sx = 0.f; sy = 0.f;
#pragma unroll
    for (int w = 0; w < 8; ++w) { sx += sRedA[w]; sy += sRedB[w]; }

    const float p = ex * __builtin_amdgcn_rcpf(sx);
    const float q = ey * __builtin_amdgcn_rcpf(sy);
    sP[tid] = p;
    sQ[tid] = q;

    // per-element marginal entropy contributions: -(p*log2(p+eps) + q*log2(q+eps))
    const float ent = -(p * __builtin_amdgcn_logf(p + EPSF) +
                        q * __builtin_amdgcn_logf(q + EPSF));
    __syncthreads();  // sP/sQ visible to all waves; sRedA/B reads done

    // ---- joint term via f32 WMMA outer-product tiles ----
    // A (16x4 f32): lanes 0-15 VGPR0 = K=0 column = p[I..I+15]; all other K zero.
    // B (4x16 f32): lanes 0-15 VGPR0 = K=0 row    = q[J..J+15]; all other K zero.
    // D (16x16 f32): lane l, reg r holds p[Ibase + r + 8*(l>=16)] * q[Jbase + (l&15)].
    //
    // All fragments preloaded into registers (unconditional ds_load +
    // v_cndmask zeroing of lanes 16-31), and the tile loops are FULLY
    // unrolled so fragment accesses are compile-time register references
    // (no M0 / v_movrels indexing, no loop-control SALU in the hot path).
    const int  l16    = lane & 15;
    const bool loHalf = (lane < 16);

    float qfrag[16];
#pragma unroll
    for (int J = 0; J < 16; ++J) {
        const float qv = sQ[(J << 4) + l16];   // unconditional ds_load
        qfrag[J] = loHalf ? qv : 0.f;          // v_cndmask
    }
    float pfrag[2];
#pragma unroll
    for (int ii = 0; ii < 2; ++ii) {
        const float pv = sP[(((wave << 1) + ii) << 4) + l16];
        pfrag[ii] = loHalf ? pv : 0.f;
    }

    float acc0 = 0.f, acc1 = 0.f, acc2 = 0.f, acc3 = 0.f;
    const v8f czero = {};

#pragma unroll
    for (int ii = 0; ii < 2; ++ii) {
        v2f a; a[0] = pfrag[ii]; a[1] = 0.f;
#pragma unroll
        for (int J = 0; J < 16; ++J) {
            v2f b; b[0] = qfrag[J]; b[1] = 0.f;
            v8f d = __builtin_amdgcn_wmma_f32_16x16x4_f32(
                false, a, false, b, (short)0, czero, false, false);
#pragma unroll
            for (int r = 0; r < 8; ++r) {
                const float t = d[r];
                const float l = __builtin_amdgcn_logf(t + EPSF);  // log2(t+eps)
                if ((r & 3) == 0)      acc0 = fmaf(t, l, acc0);
                else if ((r & 3) == 1) acc1 = fmaf(t, l, acc1);
                else if ((r & 3) == 2) acc2 = fmaf(t, l, acc2);
                else                   acc3 = fmaf(t, l, acc3);
            }
        }
    }
    const float acc = (acc0 + acc1) + (acc2 + acc3);  // = sum pq*log2(pq+eps) (partial)

    // ---- block reduction: result_row = E + S  (E - e_xy, with e_xy = -S) ----
    const float accW = waveReduceSum(acc);
    const float entW = waveReduceSum(ent);
    if (lane == 0) { sRedA[wave] = accW; sRedB[wave] = entW; }
    __syncthreads();
    if (tid == 0) {
        float S = 0.f, E = 0.f;
#pragma unroll
        for (int w = 0; w < 8; ++w) { S += sRedA[w]; E += sRedB[w]; }
        partial[row] = E + S;
    }
}

// Deterministic tree reduction of the 4096 per-row values -> scalar mean.
__global__ __launch_bounds__(256) void gmi_final_kernel(const float* __restrict__ partial,
                                                        float* __restrict__ out) {
    __shared__ float sRed[8];
    const int tid = threadIdx.x;
    float v = 0.f;
    for (int i = tid; i < D_ROWS; i += 256) v += partial[i];
    const float w = waveReduceSum(v);
    if ((tid & 31) == 0) sRed[tid >> 5] = w;
    __syncthreads();
    if (tid == 0) {
        float s = 0.f;
#pragma unroll
        for (int i = 0; i < 8; ++i) s += sRed[i];
        out[0] = s * (1.0f / (float)D_ROWS);
    }
}

extern "C" void kernel_launch(void* const* d_in, const int* in_sizes, int n_in,
                              void* d_out, int out_size, void* d_ws, size_t ws_size,
                              hipStream_t stream) {
    const float* x = (const float*)d_in[0];
    const float* y = (const float*)d_in[1];
    float* partial = (float*)d_ws;  // D_ROWS floats = 16 KB

    gmi_row_kernel<<<D_ROWS, 256, 0, stream>>>(x, y, partial);
    gmi_final_kernel<<<1, 256, 0, stream>>>(partial, (float*)d_out);
}